// GraphRegularizationModel_11098195493609
// MI455X (gfx1250) — compile-verified
//
#include <hip/hip_runtime.h>

typedef float v2f __attribute__((ext_vector_type(2)));
typedef float v8f __attribute__((ext_vector_type(8)));

#define WAVES_PER_BLOCK 8
#define TB 256

// ---------------------------------------------------------------------------
// init: prefix sums of n_node / n_edge, zero per-graph accumulators
// ---------------------------------------------------------------------------
__global__ void init_kernel(const int* __restrict__ n_node,
                            const int* __restrict__ n_edge, int G,
                            float* __restrict__ ew, float* __restrict__ ga,
                            int* __restrict__ node_cum, int* __restrict__ edge_cum) {
  int t = threadIdx.x;
  if (t < G) { ew[t] = 0.f; ga[t] = 0.f; }
  if (t == 0) {
    int cn = 0, ce = 0;
    edge_cum[0] = 0;
    for (int g = 0; g < G; ++g) {
      cn += n_node[g]; node_cum[g] = cn;
      ce += n_edge[g]; edge_cum[g + 1] = ce;
    }
  }
}

// ---------------------------------------------------------------------------
// C[M x N] = act(A[M x K] @ B[K x N] + bias), fp32 WMMA 16x16x4.
// One wave per 16x16 tile; 8 waves/block share one 16xK A tile in padded LDS.
// Grid sized so M,N,K tile counts divide exactly (M=10000, N in {512,128},
// K in {128,512}).
// ---------------------------------------------------------------------------
__global__ void gemm_relu_wmma(const float* __restrict__ A,
                               const float* __restrict__ B,
                               const float* __restrict__ bias,
                               float* __restrict__ C,
                               int N, int K, int kbits, int do_relu) {
  extern __shared__ float lds[];
  const int lstride = K + 2;                 // pad 2 floats -> conflict-free b64 reads
  const int ntiles = N >> 4;
  const int wave = threadIdx.x >> 5;
  const int lane = threadIdx.x & 31;
  const int t = blockIdx.x * WAVES_PER_BLOCK + wave;
  const int mt = t / ntiles;
  const int nt = t - mt * ntiles;
  const int mt0 = (blockIdx.x * WAVES_PER_BLOCK) / ntiles;  // same mt for all 8 waves

  // cooperative stage of the 16xK A tile (16*K is a multiple of 256: uniform trips)
  for (int idx = threadIdx.x; idx < (K << 4); idx += TB) {
    int row = idx >> kbits;
    int col = idx & (K - 1);
    lds[row * lstride + col] = A[((mt0 << 4) + row) * K + col];
  }
  __syncthreads();

  const int hf = lane >> 4;        // lanes 16-31 hold K+2,K+3 of the A/B fragments
  const int mrow = lane & 15;
  const int col = (nt << 4) + (lane & 15);
  const float* arow = lds + mrow * lstride + (hf << 1);

  v8f acc = {0.f, 0.f, 0.f, 0.f, 0.f, 0.f, 0.f, 0.f};
#pragma unroll 4
  for (int k0 = 0; k0 < K; k0 += 4) {
    v2f a, b;
    a.x = arow[k0];
    a.y = arow[k0 + 1];
    int kb = k0 + (hf << 1);
    b.x = B[kb * N + col];
    b.y = B[(kb + 1) * N + col];
    // D = A(16x4,f32) * B(4x16,f32) + C  -> v_wmma_f32_16x16x4_f32
    acc = __builtin_amdgcn_wmma_f32_16x16x4_f32(
        /*neg_a=*/false, a, /*neg_b=*/false, b,
        /*c_mod=*/(short)0, acc, /*reuse_a=*/false, /*reuse_b=*/false);
  }

  const float bv = bias[col];
#pragma unroll
  for (int r = 0; r < 8; ++r) {
    float v = acc[r] + bv;
    if (do_relu) v = fmaxf(v, 0.f);
    C[((mt << 4) + r + (hf << 3)) * N + col] = v;   // C layout: VGPR r -> row r (+8 for hi half)
  }
}

// ---------------------------------------------------------------------------
// edge phase: one wave per `epw` contiguous edges. Each lane owns a float4
// slice of the 128 features. Per-lane accumulation of w*sum(diff^2); cross-
// lane reduce + global atomic only at graph boundaries / chunk end.
// ---------------------------------------------------------------------------
__global__ void edge_kernel(const float* __restrict__ h,
                            const float* __restrict__ edges,
                            const int* __restrict__ senders,
                            const int* __restrict__ receivers,
                            const int* __restrict__ edge_cum,
                            int E, int G, int epw,
                            float* __restrict__ ew, float* __restrict__ ga) {
  const int wi = (int)((blockIdx.x * blockDim.x + threadIdx.x) >> 5);
  const int lane = threadIdx.x & 31;
  long e0l = (long)wi * epw;
  if (e0l >= E) return;
  const int e0 = (int)e0l;
  const int eEnd = (e0 + epw < E) ? (e0 + epw) : E;

  int gid = 0;
  while (gid < G - 1 && e0 >= edge_cum[gid + 1]) gid++;

  float acc = 0.f;   // per-lane partial of sum_e w * sum_f diff^2 (this graph)
  float wsum = 0.f;  // identical in every lane; lane 0's copy is flushed

  for (int e = e0; e < eEnd; ++e) {
    if (gid < G - 1 && e >= edge_cum[gid + 1]) {
      float r = acc;
      for (int m = 16; m >= 1; m >>= 1) r += __shfl_xor(r, m, 32);
      if (lane == 0) { atomicAdd(&ga[gid], r); atomicAdd(&ew[gid], wsum); }
      acc = 0.f; wsum = 0.f;
      do { gid++; } while (gid < G - 1 && e >= edge_cum[gid + 1]);
    }
    const int s = senders[e];
    const int d = receivers[e];
    const float w = edges[e];
    const float4* hs = (const float4*)(h + (long)s * 128);
    const float4* hd = (const float4*)(h + (long)d * 128);
    float4 a = hs[lane];
    float4 b = hd[lane];
    float dx = a.x - b.x, dy = a.y - b.y, dz = a.z - b.z, dw = a.w - b.w;
    acc += w * (dx * dx + dy * dy + dz * dz + dw * dw);
    wsum += w;
  }
  float r = acc;
  for (int m = 16; m >= 1; m >>= 1) r += __shfl_xor(r, m, 32);
  if (lane == 0) { atomicAdd(&ga[gid], r); atomicAdd(&ew[gid], wsum); }
}

// ---------------------------------------------------------------------------
// finalize: blocks 0..G-1 gather node_out rows; block G computes guarded mean
// ---------------------------------------------------------------------------
__global__ void finalize_kernel(const float* __restrict__ h,
                                const int* __restrict__ node_cum,
                                const float* __restrict__ ew,
                                const float* __restrict__ ga,
                                int G, float* __restrict__ out, int out_size) {
  if ((int)blockIdx.x < G) {
    const int nid = node_cum[blockIdx.x] - 1;
    out[blockIdx.x * blockDim.x + threadIdx.x] =
        h[(long)nid * blockDim.x + threadIdx.x];
  } else {
    __shared__ float vals[128];
    const int t = threadIdx.x;
    float v = 0.f;
    if (t < G) { float e = ew[t]; v = (e != 0.f) ? ga[t] / e : 0.f; }
    vals[t] = v;
    __syncthreads();
    if (t == 0) {
      float s = 0.f;
      for (int i = 0; i < G; ++i) s += vals[i];
      out[out_size - 1] = s / (float)G;
    }
  }
}

// ---------------------------------------------------------------------------
extern "C" void kernel_launch(void* const* d_in, const int* in_sizes, int n_in,
                              void* d_out, int out_size, void* d_ws, size_t ws_size,
                              hipStream_t stream) {
  const float* nodes     = (const float*)d_in[0];
  const float* edges     = (const float*)d_in[1];
  const int*   senders   = (const int*)d_in[2];
  const int*   receivers = (const int*)d_in[3];
  const int*   n_node    = (const int*)d_in[4];
  const int*   n_edge    = (const int*)d_in[5];
  const float* W1        = (const float*)d_in[6];
  const float* b1        = (const float*)d_in[7];
  const float* W2        = (const float*)d_in[8];
  const float* b2        = (const float*)d_in[9];

  const int H = in_sizes[7];              // 512
  const int D = in_sizes[9];              // 128
  const int M = in_sizes[0] / D;          // 10000 nodes
  const int E = in_sizes[2];              // 640000 edges
  const int G = in_sizes[4];              // 100 graphs
  float* out = (float*)d_out;

  // workspace layout
  char* ws = (char*)d_ws;
  size_t off = 0;
  float* hid = (float*)(ws + off); off += (size_t)M * H * sizeof(float);
  float* h   = (float*)(ws + off); off += (size_t)M * D * sizeof(float);
  float* ew  = (float*)(ws + off); off += (size_t)G * sizeof(float);
  float* ga  = (float*)(ws + off); off += (size_t)G * sizeof(float);
  int* node_cum = (int*)(ws + off); off += (size_t)G * sizeof(int);
  int* edge_cum = (int*)(ws + off); off += (size_t)(G + 1) * sizeof(int);

  init_kernel<<<1, 256, 0, stream>>>(n_node, n_edge, G, ew, ga, node_cum, edge_cum);

  // GEMM1: hid = relu(nodes @ W1 + b1)   M x H, K = D
  {
    int kbits = 31 - __builtin_clz(D);
    int blocks = (M / 16) * (H / 16) / WAVES_PER_BLOCK;
    size_t lds = 16u * (D + 2) * sizeof(float);
    gemm_relu_wmma<<<blocks, TB, lds, stream>>>(nodes, W1, b1, hid, H, D, kbits, 1);
  }
  // GEMM2: h = hid @ W2 + b2             M x D, K = H
  {
    int kbits = 31 - __builtin_clz(H);
    int blocks = (M / 16) * (D / 16) / WAVES_PER_BLOCK;
    size_t lds = 16u * (H + 2) * sizeof(float);
    gemm_relu_wmma<<<blocks, TB, lds, stream>>>(hid, W2, b2, h, D, H, kbits, 0);
  }
  // edge accumulation
  {
    int blocks = 2000;
    int waves = blocks * WAVES_PER_BLOCK;
    int epw = (E + waves - 1) / waves;    // 40
    edge_kernel<<<blocks, TB, 0, stream>>>(h, edges, senders, receivers,
                                           edge_cum, E, G, epw, ew, ga);
  }
  // node gather + loss
  finalize_kernel<<<G + 1, D, 0, stream>>>(h, node_cum, ew, ga, G, out, out_size);
}